// GrowingSignature_9861244912086
// MI455X (gfx1250) — compile-verified
//
#include <hip/hip_runtime.h>

// GrowingSignature for MI455X (gfx1250, wave32).
//
// Shapes (fixed by the harness reference):
//   x      : (B=32, L=512, F=4)  float32
//   kernel : (C=10, F=4, NF=340) float32
//   out    : (B=32, L=512, NF+1=341) float32
//
// Strategy: fused lift (WMMA f32 16x16x4, K=4 == F exactly) + per-channel
// sequential scan, streaming results to the output. Store-BW bound (~22 MB).

typedef __attribute__((ext_vector_type(2))) float v2f;
typedef __attribute__((ext_vector_type(8))) float v8f;

#define BATCH 32
#define LOUT  512
#define LSEQ  511      // L-1 increments
#define FEAT  4
#define NCOMP 10       // num_levels*(num_levels+1)/2
#define NF    340      // functionals
#define NCH   341      // NF + constant-1 channel
#define COLS  32       // functionals handled per wave (two 16-wide WMMA tiles)
#define CHUNK 16       // timesteps per WMMA tile

__global__ __launch_bounds__(32)
void sig_scan_kernel(const float* __restrict__ x,
                     const float* __restrict__ kern,
                     float* __restrict__ out)
{
    // [component][row(t)][col(nf)] tiles; conflict-free row-major reads/writes.
    __shared__ float lds[NCOMP * CHUNK * COLS];   // 20 KB

    const int lane = threadIdx.x;        // 0..31
    const int b    = blockIdx.x;         // batch
    const int nf0  = blockIdx.y * COLS;  // first functional of this group

    const int half = lane >> 4;          // selects K-pair per ISA A/B layout
    const int l15  = lane & 15;          // row (A) / col (B) within fragment
    const int kb   = half * 2;           // K base: lanes 0-15 -> K{0,1}, 16-31 -> K{2,3}

    // ---- B fragments: kernel[c][K][nf], loop-invariant over time ----------
    // B 4x16 layout (ISA 7.12.2): v0 = K=kb (lanes by column), v1 = K=kb+1.
    v2f blo[NCOMP], bhi[NCOMP];
    #pragma unroll
    for (int c = 0; c < NCOMP; ++c) {
        const int c0 = nf0 + l15;        // low 16-col tile
        const int c1 = nf0 + 16 + l15;   // high 16-col tile
        v2f t0; t0.x = 0.f; t0.y = 0.f;
        v2f t1; t1.x = 0.f; t1.y = 0.f;
        if (c0 < NF) {
            t0.x = kern[(c * FEAT + kb    ) * NF + c0];
            t0.y = kern[(c * FEAT + kb + 1) * NF + c0];
        }
        if (c1 < NF) {
            t1.x = kern[(c * FEAT + kb    ) * NF + c1];
            t1.y = kern[(c * FEAT + kb + 1) * NF + c1];
        }
        blo[c] = t0;
        bhi[c] = t1;
    }

    // ---- scan state: 6 exclusive-prefix accumulators + running output ----
    float p21 = 0.f, p31 = 0.f, p32 = 0.f;
    float p41 = 0.f, p42 = 0.f, p43 = 0.f;
    float tot = 0.f;

    const float* xb   = x   + (size_t)b * LOUT * FEAT;
    float*       outb = out + (size_t)b * LOUT * NCH;

    const int col = lane;           // functional column owned in the scan
    const int nf  = nf0 + col;

    for (int t0 = 0; t0 < LSEQ; t0 += CHUNK) {
        // ---- A fragment: seq rows t0..t0+15, features kb..kb+1 ------------
        const int row = t0 + l15;
        v2f a; a.x = 0.f; a.y = 0.f;
        if (row < LSEQ) {
            const float x0a = xb[ row      * FEAT + kb];
            const float x0b = xb[ row      * FEAT + kb + 1];
            const float x1a = xb[(row + 1) * FEAT + kb];
            const float x1b = xb[(row + 1) * FEAT + kb + 1];
            a.x = x1a - x0a;
            a.y = x1b - x0b;
        }

        // ---- lift: M[c] tiles via v_wmma_f32_16x16x4_f32 ------------------
        #pragma unroll
        for (int c = 0; c < NCOMP; ++c) {
            v8f z = {};
            v8f dlo = __builtin_amdgcn_wmma_f32_16x16x4_f32(
                false, a, false, blo[c], (short)0, z, false, false);
            v8f dhi = __builtin_amdgcn_wmma_f32_16x16x4_f32(
                false, a, false, bhi[c], (short)0, z, false, false);
            #pragma unroll
            for (int v = 0; v < 8; ++v) {
                const int r = v + half * 8;          // D layout: lane half -> M+8
                lds[(c * CHUNK + r) * COLS + l15]      = dlo[v];
                lds[(c * CHUNK + r) * COLS + 16 + l15] = dhi[v];
            }
        }
        __syncthreads();   // single-wave WG: barrier == NOP, forces s_wait_dscnt

        // ---- sequential recurrence along time, one lane per functional ----
        for (int r = 0; r < CHUNK; ++r) {
            const int t = t0 + r;          // uniform across the wave
            if (t >= LSEQ) break;
            const float m0 = lds[(0 * CHUNK + r) * COLS + col];
            const float m1 = lds[(1 * CHUNK + r) * COLS + col];
            const float m2 = lds[(2 * CHUNK + r) * COLS + col];
            const float m3 = lds[(3 * CHUNK + r) * COLS + col];
            const float m4 = lds[(4 * CHUNK + r) * COLS + col];
            const float m5 = lds[(5 * CHUNK + r) * COLS + col];
            const float m6 = lds[(6 * CHUNK + r) * COLS + col];
            const float m7 = lds[(7 * CHUNK + r) * COLS + col];
            const float m8 = lds[(8 * CHUNK + r) * COLS + col];
            const float m9 = lds[(9 * CHUNK + r) * COLS + col];

            // level 1
            const float r1 = m0;
            // level 2: R = M2 * exclusive_cumsum(M1)
            const float r2 = m2 * p21;  p21 += m1;
            // level 3: R' = M4 * ex(M3); R = M5 * ex(R')
            const float s3 = m4 * p31;  p31 += m3;
            const float r3 = m5 * p32;  p32 += s3;
            // level 4: chain of three stages
            const float s4 = m7 * p41;  p41 += m6;
            const float s5 = m8 * p42;  p42 += s4;
            const float r4 = m9 * p43;  p43 += s5;

            tot += r1 + r2 + r3 + r4;
            if (nf < NF)
                outb[(size_t)(t + 1) * NCH + 1 + nf] = tot;
        }
        __syncthreads();   // protect LDS against next chunk's overwrites
    }
}

// Fill constant-1 channel (all t) and the zero t=0 row (channels 1..340).
__global__ void sig_init_kernel(float* __restrict__ out)
{
    const int idx = blockIdx.x * blockDim.x + threadIdx.x;  // over B*L
    if (idx < BATCH * LOUT) {
        const int b = idx / LOUT;
        const int t = idx % LOUT;
        out[(size_t)(b * LOUT + t) * NCH] = 1.0f;
        if (t < NF)
            out[(size_t)b * LOUT * NCH + 1 + t] = 0.0f;
    }
}

extern "C" void kernel_launch(void* const* d_in, const int* in_sizes, int n_in,
                              void* d_out, int out_size, void* d_ws, size_t ws_size,
                              hipStream_t stream)
{
    (void)in_sizes; (void)n_in; (void)out_size; (void)d_ws; (void)ws_size;
    const float* x    = (const float*)d_in[0];
    const float* kern = (const float*)d_in[1];
    float*       out  = (float*)d_out;

    sig_init_kernel<<<(BATCH * LOUT + 255) / 256, 256, 0, stream>>>(out);

    dim3 grid(BATCH, (NF + COLS - 1) / COLS);   // 32 x 11 single-wave workgroups
    sig_scan_kernel<<<grid, 32, 0, stream>>>(x, kern, out);
}